// PerceptualPooling_33449205301376
// MI455X (gfx1250) — compile-verified
//
#include <hip/hip_runtime.h>
#include <math.h>

#define MAP_SIZE 224
#define NPTS     8192
#define NB       4

typedef __attribute__((ext_vector_type(2))) float v2f;
typedef __attribute__((ext_vector_type(8))) float v8f;

// ---------------------------------------------------------------------------
// Kernel 1: project points with V_WMMA_F32_16X16X4_F32.
//   D = A(16x4) * B(4x16),  A = trans_mat^T (3x4 zero-padded to 16x4),
//   B = 16 homogeneous points as columns.
//   D[j][n] = sum_k trans[k][j] * homo[n][k] = pc_xyz[n][j]
//   => lane n (n<16) holds x' = d[0], y' = d[1], z' = d[2] of point n.
// One wave (32 threads) per 16 points. EXEC is all-ones at the WMMA.
// ---------------------------------------------------------------------------
__global__ __launch_bounds__(32)
void project_points_wmma(const float* __restrict__ pc,      // (B, N, 3)
                         const float* __restrict__ trans,   // (B, 4, 3)
                         float* __restrict__ xs,            // (B*N)
                         float* __restrict__ ys) {          // (B*N)
    const int lane  = threadIdx.x;            // 0..31
    const int p0    = blockIdx.x * 16;        // first flat point of this tile
    const int b     = p0 / NPTS;              // N % 16 == 0 -> tile in one batch
    const int m     = lane & 15;
    const int khalf = lane >> 4;              // 0: K=0,1   1: K=2,3

    // A-matrix (16x4 f32): lanes 0-15 -> VGPR0=K0, VGPR1=K1; lanes 16-31 -> K2,K3
    // A[m][k] = trans[b][k][m] for m<3 else 0
    v2f a;
    {
        const int kb = khalf * 2;
        const float* tb = trans + b * 12;
        a.x = (m < 3) ? tb[kb * 3 + m]       : 0.0f;
        a.y = (m < 3) ? tb[(kb + 1) * 3 + m] : 0.0f;
    }

    // B-matrix (4x16 f32): lanes 0-15 hold rows K=0,1 (x,y); lanes 16-31 rows K=2,3 (z,1)
    v2f bm;
    {
        const int   fp  = p0 + m;             // flat point index (b*N + n)
        const float* pp = pc + (size_t)fp * 3;
        bm.x = khalf ? pp[2] : pp[0];
        bm.y = khalf ? 1.0f  : pp[1];
    }

    v8f c = {};
    // 8 args: (neg_a, A, neg_b, B, c_mod, C, reuse_a, reuse_b)
    v8f d = __builtin_amdgcn_wmma_f32_16x16x4_f32(
        false, a, false, bm, (short)0, c, false, false);

    if (lane < 16) {
        const int fp = p0 + lane;
        const float xw = d[0];
        const float yw = d[1];
        const float zw = d[2] + 1e-8f;
        float px = xw / zw;
        float py = yw / zw;
        px = fminf(fmaxf(px, 0.0f), (float)(MAP_SIZE - 1));
        py = fminf(fmaxf(py, 0.0f), (float)(MAP_SIZE - 1));
        xs[fp] = px;
        ys[fp] = py;
    }
}

// ---------------------------------------------------------------------------
// Kernel 2: fused (resize-bilinear o grid-sample-bilinear) as a separable
// 4x4 stencil on the ORIGINAL feature map -- the 224x224 intermediate is
// never materialized. One thread per output element (b, c, n); indices and
// weights depend only on (b, n, map) and are recomputed in-register (the 16
// gathers dominate; sources fit in the 192 MB L2).
// ---------------------------------------------------------------------------
__device__ __forceinline__ void resize_tap(int g, float scale, int H,
                                           int& i0, int& i1, float& w) {
    const float cc = (float)g * scale;   // matches jnp: arange*scale in f32
    i0 = (int)floorf(cc);
    i1 = min(i0 + 1, H - 1);
    w  = cc - (float)i0;
}

__global__ __launch_bounds__(256)
void sample_map(const float* __restrict__ src,   // (B, C, H, H)
                const float* __restrict__ xs,
                const float* __restrict__ ys,
                float* __restrict__ out,         // (B, C, 1, N) flat
                int C, int H) {
    const int tid   = blockIdx.x * blockDim.x + threadIdx.x;
    const int total = NB * C * NPTS;
    if (tid >= total) return;

    const int n = tid % NPTS;
    const int c = (tid / NPTS) % C;
    const int b = tid / (NPTS * C);

    const int   fp = b * NPTS + n;
    const float px = xs[fp];
    const float py = ys[fp];

    const float scale = (float)(H - 1) / (float)(MAP_SIZE - 1);

    // grid-sample corners in 224-space
    const int   gx0 = (int)floorf(px);
    const int   gx1 = min(gx0 + 1, MAP_SIZE - 1);
    const float fx  = px - (float)gx0;
    const int   gy0 = (int)floorf(py);
    const int   gy1 = min(gy0 + 1, MAP_SIZE - 1);
    const float fy  = py - (float)gy0;

    // compose with resize weights -> 4 y-taps, 4 x-taps
    int   yi[4], xi[4];
    float vy[4], vx[4];
    {
        int i0, i1; float w;
        resize_tap(gy0, scale, H, i0, i1, w);
        yi[0] = i0; yi[1] = i1; vy[0] = (1.0f - fy) * (1.0f - w); vy[1] = (1.0f - fy) * w;
        resize_tap(gy1, scale, H, i0, i1, w);
        yi[2] = i0; yi[3] = i1; vy[2] = fy * (1.0f - w);          vy[3] = fy * w;
        resize_tap(gx0, scale, H, i0, i1, w);
        xi[0] = i0; xi[1] = i1; vx[0] = (1.0f - fx) * (1.0f - w); vx[1] = (1.0f - fx) * w;
        resize_tap(gx1, scale, H, i0, i1, w);
        xi[2] = i0; xi[3] = i1; vx[2] = fx * (1.0f - w);          vx[3] = fx * w;
    }

    const float* plane = src + ((size_t)(b * C + c)) * H * H;
    float acc = 0.0f;
#pragma unroll
    for (int p = 0; p < 4; ++p) {
        const float* row = plane + (size_t)yi[p] * H;
        const float r = vx[0] * row[xi[0]] + vx[1] * row[xi[1]] +
                        vx[2] * row[xi[2]] + vx[3] * row[xi[3]];
        acc = fmaf(vy[p], r, acc);
    }
    out[tid] = acc;
}

// ---------------------------------------------------------------------------
// Launch. Inputs: x1,x2,x3,x4,x5,pc,trans_mat,sdf_value (sdf unused).
// Output: outf1..outf5 concatenated flat, each (B, C_i, 1, N), f32.
// Workspace: xs (B*N f32) + ys (B*N f32) = 256 KB.
// ---------------------------------------------------------------------------
extern "C" void kernel_launch(void* const* d_in, const int* in_sizes, int n_in,
                              void* d_out, int out_size, void* d_ws, size_t ws_size,
                              hipStream_t stream) {
    const float* x[5];
    for (int i = 0; i < 5; ++i) x[i] = (const float*)d_in[i];
    const float* pc    = (const float*)d_in[5];
    const float* trans = (const float*)d_in[6];

    float* xs = (float*)d_ws;
    float* ys = xs + NB * NPTS;

    // 1) WMMA projection: 2048 tiles of 16 points, one wave each
    {
        const int tiles = (NB * NPTS) / 16;
        project_points_wmma<<<tiles, 32, 0, stream>>>(pc, trans, xs, ys);
    }

    // 2) Fused gather-sample per pyramid level
    static const int Cs[5] = { 64, 128, 256, 512, 512 };
    static const int Hs[5] = { 224, 112, 56, 28, 14 };

    float* out = (float*)d_out;
    size_t off = 0;
    for (int m = 0; m < 5; ++m) {
        const int total  = NB * Cs[m] * NPTS;
        const int blocks = (total + 255) / 256;
        sample_map<<<blocks, 256, 0, stream>>>(x[m], xs, ys, out + off, Cs[m], Hs[m]);
        off += (size_t)total;
    }
}